// ROIPool_81784767250573
// MI455X (gfx1250) — compile-verified
//
#include <hip/hip_runtime.h>
#include <hip/hip_bf16.h>
#include <math.h>

#define NUM_GRAPHS 1024
#define NPG        400
#define CH         128
#define KSEL       200
#define N_NODES    (NUM_GRAPHS * NPG)   // 409600

// d_out layout (flat, return order): x_pooled | batch_pooled | scores | perm
#define X_POOLED_N ((size_t)NUM_GRAPHS * KSEL * CH)   // 26214400
#define POOLED_N   ((size_t)NUM_GRAPHS * KSEL)        // 204800

typedef __attribute__((ext_vector_type(2))) float v2f;
typedef __attribute__((ext_vector_type(8))) float v8f;

#define LDS_STRIDE 136   // 2*136 % 64 == 16 -> lane halves use disjoint LDS bank sets

// ---------------------------------------------------------------------------
// Kernel 1: scores = tanh(x @ W1 + b1) @ w2 + b2, via f32 WMMA (16x16x4).
// 8 waves/WG, each wave computes scores for one 16-node tile.
// ---------------------------------------------------------------------------
extern "C" __global__ void __launch_bounds__(256)
score_kernel(const float* __restrict__ x,
             const float* __restrict__ W1,
             const float* __restrict__ b1,
             const float* __restrict__ w2,
             const float* __restrict__ b2,
             float* __restrict__ scores_out)
{
    extern __shared__ float sW1[];   // CH rows x LDS_STRIDE words = 69632 B

    const int tid = threadIdx.x;
    // cooperative stage of W1 into LDS (row-padded)
    for (int i = tid; i < CH * CH; i += 256) {
        int r = i >> 7, c = i & 127;
        sW1[r * LDS_STRIDE + c] = W1[i];
    }
    __syncthreads();

    const int lane = tid & 31;
    const int wave = tid >> 5;
    const int half = lane >> 4;      // 0: lanes 0-15, 1: lanes 16-31
    const int l16  = lane & 15;
    const int koff = half * 2;       // A-layout: K offset 0 or 2
    const int node_base = blockIdx.x * 128 + wave * 16;

    // Preload all 32 A fragments (16x4 f32): lane l16 = row M, float2 = {K, K+1}
    const float* xrow = x + (size_t)(node_base + l16) * CH + koff;
    v2f a[32];
#pragma unroll
    for (int kk = 0; kk < 32; ++kk)
        a[kk] = *(const v2f*)(xrow + kk * 4);   // 8B-aligned (even float offset)

    float s[8];
#pragma unroll
    for (int r = 0; r < 8; ++r) s[r] = 0.0f;

    for (int j = 0; j < 8; ++j) {                 // 8 hidden-column tiles of 16
        const int ncol = j * 16 + l16;            // this lane's N column
        v8f acc = {};
#pragma unroll
        for (int kk = 0; kk < 32; ++kk) {
            // B fragment (4x16 f32): b = {W1[k+koff][ncol], W1[k+koff+1][ncol]}
            const float* wp = &sW1[(kk * 4 + koff) * LDS_STRIDE + ncol];
            v2f bf;
            bf.x = wp[0];
            bf.y = wp[LDS_STRIDE];
            acc = __builtin_amdgcn_wmma_f32_16x16x4_f32(
                /*neg_a=*/false, a[kk], /*neg_b=*/false, bf,
                /*c_mod=*/(short)0, acc, /*reuse_a=*/false, /*reuse_b=*/false);
        }
        // fused epilogue: s[m] += w2[n] * tanh(h[m][n] + b1[n])
        const float b1n = b1[ncol];
        const float w2n = w2[ncol];
#pragma unroll
        for (int r = 0; r < 8; ++r)
            s[r] += w2n * tanhf(acc[r] + b1n);
    }

    // reduce over the 16 lanes of each half (masks stay within a half)
#pragma unroll
    for (int r = 0; r < 8; ++r) {
        float v = s[r];
        v += __shfl_xor(v, 1);
        v += __shfl_xor(v, 2);
        v += __shfl_xor(v, 4);
        v += __shfl_xor(v, 8);
        s[r] = v;
    }
    if (l16 == 0) {
        const float bb = b2[0];
        const int m = node_base + half * 8;   // C/D layout: M = r + 8*half
#pragma unroll
        for (int r = 0; r < 8; ++r)
            scores_out[m + r] = s[r] + bb;
    }
}

// ---------------------------------------------------------------------------
// Kernel 2: per-graph stable descending top-200 by rank counting, then
// gather rows scaled by sigmoid(score); emit batch & perm as floats.
// ---------------------------------------------------------------------------
extern "C" __global__ void __launch_bounds__(256)
topk_gather_kernel(const float* __restrict__ x,
                   const float* __restrict__ scores,
                   float* __restrict__ out_x,
                   float* __restrict__ out_batch,
                   float* __restrict__ out_perm)
{
    __shared__ float sc[NPG];
    __shared__ int   sel[KSEL];
    __shared__ float sgate[KSEL];

    const int g   = blockIdx.x;
    const int tid = threadIdx.x;

    for (int i = tid; i < NPG; i += 256)
        sc[i] = scores[(size_t)g * NPG + i];
    __syncthreads();

    for (int i = tid; i < NPG; i += 256) {
        const float si = sc[i];
        int rank = 0;
        for (int j = 0; j < NPG; ++j) {
            const float sj = sc[j];
            rank += (sj > si) || ((sj == si) && (j < i));  // stable descending
        }
        if (rank < KSEL) {
            sel[rank]   = i;
            sgate[rank] = 1.0f / (1.0f + expf(-si));
        }
    }
    __syncthreads();

    for (int p = tid; p < KSEL; p += 256) {
        out_batch[(size_t)g * KSEL + p] = (float)g;
        out_perm [(size_t)g * KSEL + p] = (float)(g * NPG + sel[p]);
    }

    const float4* x4 = (const float4*)x;
    float4*       o4 = (float4*)out_x;
    for (int t = tid; t < KSEL * (CH / 4); t += 256) {
        const int p = t >> 5;       // CH/4 == 32 float4 per row
        const int q = t & 31;
        const int node = sel[p];
        const float gg = sgate[p];
        float4 v = x4[((size_t)g * NPG + node) * (CH / 4) + q];
        v.x *= gg; v.y *= gg; v.z *= gg; v.w *= gg;
        o4[((size_t)g * KSEL + p) * (CH / 4) + q] = v;
    }
}

// ---------------------------------------------------------------------------
extern "C" void kernel_launch(void* const* d_in, const int* in_sizes, int n_in,
                              void* d_out, int out_size, void* d_ws, size_t ws_size,
                              hipStream_t stream) {
    const float* x  = (const float*)d_in[0];
    // d_in[1] = batch (int64) — implied by graph id, unused
    const float* W1 = (const float*)d_in[2];
    const float* b1 = (const float*)d_in[3];
    const float* w2 = (const float*)d_in[4];
    const float* b2 = (const float*)d_in[5];

    float* out       = (float*)d_out;
    float* out_x     = out;                        // 26,214,400
    float* out_batch = out + X_POOLED_N;           //    204,800
    float* out_sc    = out_batch + POOLED_N;       //    409,600
    float* out_perm  = out_sc + (size_t)N_NODES;   //    204,800

    const size_t lds_bytes = (size_t)CH * LDS_STRIDE * sizeof(float);  // 69632
    score_kernel<<<N_NODES / 128, 256, lds_bytes, stream>>>(
        x, W1, b1, w2, b2, out_sc);
    topk_gather_kernel<<<NUM_GRAPHS, 256, 0, stream>>>(
        x, out_sc, out_x, out_batch, out_perm);
}